// PowerSpectrum_33423435497818
// MI455X (gfx1250) — compile-verified
//
#include <hip/hip_runtime.h>
#include <math.h>

// CDNA5 / gfx1250 wave32 WMMA types
typedef float v2f __attribute__((ext_vector_type(2)));
typedef float v4f __attribute__((ext_vector_type(4)));
typedef float v8f __attribute__((ext_vector_type(8)));

#define J_ENV   30000
#define A_DIM   32
#define NPAD    544               // padded packed width per l (528 real + 16 pad)
#define NL      7                 // lmax+1
#define ROWLEN  (NL * NPAD)       // 3808 floats per environment (15232 B, 16B-aligned)
#define NV4     (ROWLEN / 4)      // 952 float4s per row
#define WAVES_PER_BLOCK 4
#define BLOCK_THREADS (WAVES_PER_BLOCK * 32)

// Packed upper-tri row base: p(i,j) = triu_base(i) + j  for j >= i.
__device__ __forceinline__ constexpr int triu_base(int i) { return (i * (63 - i)) / 2; }

__global__ __launch_bounds__(BLOCK_THREADS, 1)
void powerspectrum_wmma_kernel(const float* __restrict__ se0,
                               const float* __restrict__ se1,
                               const float* __restrict__ se2,
                               const float* __restrict__ se3,
                               const float* __restrict__ se4,
                               const float* __restrict__ se5,
                               const float* __restrict__ se6,
                               float* __restrict__ out)
{
    __shared__ __align__(16) float lds[WAVES_PER_BLOCK * ROWLEN];

    const int lane = threadIdx.x & 31;
    const int wave = threadIdx.x >> 5;
    const int env  = blockIdx.x * WAVES_PER_BLOCK + wave;
    if (env >= J_ENV) return;  // uniform per wave

    float* buf = lds + wave * ROWLEN;

    const float* se[NL] = {se0, se1, se2, se3, se4, se5, se6};

    const int half = lane >> 4;   // 0: lanes 0-15, 1: lanes 16-31
    const int l15  = lane & 15;
    const int row0 = l15;         // row block 0 (rows 0..15)
    const int row1 = l15 + 16;    // row block 1 (rows 16..31)
    const float SQRT2 = 1.41421356237309515f;

    // ---- Loop-invariant scatter state (depends on lane/v only, NOT on l) ----
    // Invalid (lower-triangle) elements are steered into the pad slots
    // (528..543) of the l=0 block; the per-l displacement l*NPAD is a DS
    // immediate offset, so the trash lands in that l's own pad slots, which
    // are re-zeroed after the scatter.
    float* const pad = buf + 528 + l15;
    float* addr00[8];
    float* addr01[8];
    float* addr11[8];
    float  fac2[8];
    #pragma unroll
    for (int v = 0; v < 8; ++v) {
        const int i    = half ? (v + 8) : v;                           // row index
        const int f00  = half ? triu_base(v + 8)  : triu_base(v);      // const pair
        const int f11  = half ? triu_base(v + 24) : triu_base(v + 16); // const pair
        const bool valid = (l15 >= i);
        addr00[v] = valid ? (buf + f00 + l15)      : pad;
        addr01[v] = buf + f00 + l15 + 16;                              // always valid
        addr11[v] = valid ? (buf + f11 + l15 + 16) : pad;
        fac2[v]   = (l15 == i) ? 1.0f : SQRT2;                         // diag factor
    }

    #pragma unroll
    for (int l = 0; l < NL; ++l) {
        const int   K   = 2 * l + 1;
        const float rsl = rsqrtf((float)K);   // 1/sqrt(2l+1), constant-folded
        const float* base = se[l] + (size_t)env * (A_DIM * K);

        // One prefetch covers the next env's whole 32xK block (row-spread lanes).
        if (env + WAVES_PER_BLOCK < J_ENV)
            __builtin_prefetch(base + A_DIM * K * WAVES_PER_BLOCK + lane * K, 0, 1);

        v8f acc00 = {}, acc01 = {}, acc11 = {};

        const int nk = (K + 3) >> 2;
        #pragma unroll
        for (int kb = 0; kb < nk; ++kb) {
            // A-fragment 16x4 f32 layout: k = vgpr + 2*(lane>=16).
            // For C*C^T the same registers serve as both the A and B operand.
            // Branch-free OOB handling: clamp address in-bounds, cndmask to 0.
            const int  k0  = kb * 4 + 2 * half;
            const bool ok0 = (k0 < K);
            const bool ok1 = (k0 + 1 < K);
            const int  kc0 = ok0 ? k0 : 0;
            const int  kc1 = ok1 ? (k0 + 1) : 0;

            const float a0 = base[row0 * K + kc0];
            const float a1 = base[row0 * K + kc1];
            const float b0 = base[row1 * K + kc0];
            const float b1 = base[row1 * K + kc1];

            v2f fa, fb, faS, fbS;
            fa.x = ok0 ? a0 : 0.0f;  fa.y = ok1 ? a1 : 0.0f;
            fb.x = ok0 ? b0 : 0.0f;  fb.y = ok1 ? b1 : 0.0f;
            // Fold 1/sqrt(2l+1) into the A operand: acc = rsl * (C C^T).
            faS.x = fa.x * rsl;  faS.y = fa.y * rsl;
            fbS.x = fb.x * rsl;  fbS.y = fb.y * rsl;

            acc00 = __builtin_amdgcn_wmma_f32_16x16x4_f32(false, faS, false, fa,
                                                          (short)0, acc00, false, false);
            acc01 = __builtin_amdgcn_wmma_f32_16x16x4_f32(false, faS, false, fb,
                                                          (short)0, acc01, false, false);
            acc11 = __builtin_amdgcn_wmma_f32_16x16x4_f32(false, fbS, false, fb,
                                                          (short)0, acc11, false, false);
        }

        // Scatter: one v_mul + one ds_store (immediate offset l*2176B) per element.
        #pragma unroll
        for (int v = 0; v < 8; ++v) {
            addr00[v][l * NPAD] = acc00[v] * fac2[v];
            addr01[v][l * NPAD] = acc01[v] * SQRT2;   // never diagonal
            addr11[v][l * NPAD] = acc11[v] * fac2[v];
        }
    }

    // Re-zero the pad slots (they collected the steered trash writes).
    // Both halves write the same slot — harmless duplicate zero store.
    #pragma unroll
    for (int l = 0; l < NL; ++l)
        pad[l * NPAD] = 0.0f;

    // Row L2 norm: vectorized per-lane partial + wave32 shuffle reduction.
    v4f* bufv = (v4f*)buf;
    float s = 0.0f;
    for (int t = lane; t < NV4; t += 32) {
        const v4f x = bufv[t];
        s = fmaf(x.x, x.x, fmaf(x.y, x.y, fmaf(x.z, x.z, fmaf(x.w, x.w, s))));
    }
    #pragma unroll
    for (int off = 16; off > 0; off >>= 1)
        s += __shfl_xor(s, off, 32);

    const float inv = 1.0f / fmaxf(sqrtf(s), 1e-12f);

    // Coalesced b128 normalized store (457 MB stream — the kernel's roofline).
    v4f* outv = (v4f*)(out + (size_t)env * ROWLEN);
    for (int t = lane; t < NV4; t += 32) {
        v4f x = bufv[t];
        x.x *= inv; x.y *= inv; x.z *= inv; x.w *= inv;
        outv[t] = x;
    }
}

extern "C" void kernel_launch(void* const* d_in, const int* in_sizes, int n_in,
                              void* d_out, int out_size, void* d_ws, size_t ws_size,
                              hipStream_t stream) {
    (void)in_sizes; (void)n_in; (void)out_size; (void)d_ws; (void)ws_size;
    const float* se0 = (const float*)d_in[0];
    const float* se1 = (const float*)d_in[1];
    const float* se2 = (const float*)d_in[2];
    const float* se3 = (const float*)d_in[3];
    const float* se4 = (const float*)d_in[4];
    const float* se5 = (const float*)d_in[5];
    const float* se6 = (const float*)d_in[6];
    float* out = (float*)d_out;

    const int grid = (J_ENV + WAVES_PER_BLOCK - 1) / WAVES_PER_BLOCK;  // 7500
    powerspectrum_wmma_kernel<<<grid, BLOCK_THREADS, 0, stream>>>(
        se0, se1, se2, se3, se4, se5, se6, out);
}